// YoloLoss_78030965834199
// MI455X (gfx1250) — compile-verified
//
#include <hip/hip_runtime.h>
#include <hip/hip_bf16.h>

typedef __attribute__((ext_vector_type(2))) float v2f;
typedef __attribute__((ext_vector_type(8))) float v8f;

#define SGRID 7
#define CELLS 49      // 7*7
#define NCH   30      // 5*P + C
#define NBATCH 16384
#define NCLASS 20

// workspace layout (floats)
#define WS_COORD 0      // [49]
#define WS_CONF  49     // [49]
#define WS_CLS   98     // [49]
#define WS_A     152    // [52] padded A vector
#define WS_CB    204    // constB scalar
#define WS_HAS   256    // [NBATCH*CELLS] compacted has
#define WS_HDR   256    // floats to zero each launch

// ---------------------------------------------------------------- init
__global__ void yolo_init(float* __restrict__ ws) {
    int t = threadIdx.x;
    if (t < WS_HDR) ws[t] = 0.0f;
}

// ---------------------------------------------------------------- pass 1: batch reduction
__global__ void yolo_reduce(const float* __restrict__ out,
                            const float* __restrict__ tgt,
                            const int* __restrict__ imw,
                            const int* __restrict__ imh,
                            float* __restrict__ ws) {
    __shared__ float bins[3 * CELLS];
    const int tid = threadIdx.x;
    for (int k = tid; k < 3 * CELLS; k += blockDim.x) bins[k] = 0.0f;
    __syncthreads();

    const int   g  = blockIdx.x * blockDim.x + tid;   // cell index (exact grid)
    const float W  = (float)imw[0];
    const float H  = (float)imh[0];
    const float cw = W / (float)SGRID;
    const float ch = H / (float)SGRID;

    const int b  = g / CELLS;
    const int ij = g - b * CELLS;
    const int i  = ij / SGRID;
    const int j  = ij - i * SGRID;

    const float* t = tgt + (size_t)g * NCH;
    const float* o = out + (size_t)g * NCH;

    // target box
    const float t0 = t[0], t1 = t[1];
    const float tw = t[2] * W, th = t[3] * H;
    const float has = t[4];
    const float tcx = ((float)j + t0) * cw;
    const float tcy = ((float)i + t1) * ch;
    const float tx0 = tcx - tw * 0.5f, ty0 = tcy - th * 0.5f;
    const float tx1 = tcx + tw * 0.5f, ty1 = tcy + th * 0.5f;
    const float area_t = tw * th;

    // IoU + conf per predictor (conf channel per reference is o[5p+4])
    float iou_p[2], conf_p[2];
#pragma unroll
    for (int p = 0; p < 2; ++p) {
        const float px = o[5 * p + 0], py = o[5 * p + 1];
        const float pw = o[5 * p + 3] * W, ph = o[5 * p + 4] * H;
        const float pcx = ((float)j + px) * cw;
        const float pcy = ((float)i + py) * ch;
        const float px0 = pcx - pw * 0.5f, py0 = pcy - ph * 0.5f;
        const float px1 = pcx + pw * 0.5f, py1 = pcy + ph * 0.5f;
        float iw = fminf(px1, tx1) - fmaxf(px0, tx0); iw = fmaxf(iw, 0.0f);
        float ih = fminf(py1, ty1) - fmaxf(py0, ty0); ih = fmaxf(ih, 0.0f);
        const float inter = iw * ih;
        iou_p[p]  = inter / (pw * ph + area_t - inter + 1e-9f);
        conf_p[p] = o[5 * p + 4];
    }
    // argmax over reversed axis -> ties favor last predictor
    const float best_conf = (iou_p[1] >= iou_p[0]) ? conf_p[1] : conf_p[0];

    // coord / conf losses use last predictor (L = P-1 = 1)
    const float px = o[5], py = o[6];
    const float pw = o[8] * W, ph = o[9] * H;
    const float pconf = o[9];
    const float dx = t0 - px, dy = t1 - py;
    const float dw = sqrtf(tw) - sqrtf(pw);
    const float dh = sqrtf(th) - sqrtf(ph);
    const float coord = dx * dx + dy * dy + dw * dw + dh * dh;
    const float dc = has - pconf;           // t_conf == target[...,4]
    const float confm = dc * dc;

    float cls = 0.0f;
#pragma unroll
    for (int c = 0; c < NCLASS; ++c) {
        const float d = t[10 + c] - o[10 + c] * best_conf;
        cls = fmaf(d, d, cls);
    }

    // compact has for the WMMA matvec pass
    ws[WS_HAS + g] = has;

    atomicAdd(&bins[ij],             coord);
    atomicAdd(&bins[CELLS + ij],     confm);
    atomicAdd(&bins[2 * CELLS + ij], cls);
    __syncthreads();
    for (int k = tid; k < 3 * CELLS; k += blockDim.x)
        atomicAdd(&ws[k], bins[k]);     // ws[0..146] matches bins layout
}

// ---------------------------------------------------------------- finalize: A[52], constB
__global__ void yolo_finalize(float* __restrict__ ws) {
    __shared__ float red[64];
    const int tid = threadIdx.x;      // blockDim = 64
    float a = 0.0f, bsum = 0.0f;
    if (tid < CELLS) {
        const float coord = ws[WS_COORD + tid];
        const float conf  = ws[WS_CONF  + tid];
        const float cls   = ws[WS_CLS   + tid];
        a    = 5.0f * coord + 0.5f * conf;
        bsum = 0.5f * conf + cls;
    }
    if (tid < 52) ws[WS_A + tid] = (tid < CELLS) ? a : 0.0f;
    red[tid] = bsum;
    __syncthreads();
    for (int s = 32; s > 0; s >>= 1) {
        if (tid < s) red[tid] += red[tid + s];
        __syncthreads();
    }
    if (tid == 0) ws[WS_CB] = red[0];
}

// ---------------------------------------------------------------- pass 2: loss[b] = has[b,:]·A + constB via WMMA f32 16x16x4
__global__ void yolo_dot_wmma(const float* __restrict__ ws,
                              float* __restrict__ loss) {
    const int lane = threadIdx.x & 31;
    const int wave = (blockIdx.x * blockDim.x + threadIdx.x) >> 5;
    const int b0   = wave * 16;                 // 16 batches per wave
    const float* __restrict__ Avec = ws + WS_A;
    const float* __restrict__ hasv = ws + WS_HAS;

    const int half = lane >> 4;                 // lane group 0..15 vs 16..31
    const int m    = lane & 15;                 // A-matrix row (batch within tile)

    v8f acc = {};
#pragma unroll
    for (int k0 = 0; k0 < 52; k0 += 4) {
        // 32-bit A 16x4 layout: lanes 0-15 hold K={k0,k0+1}, lanes 16-31 hold K={k0+2,k0+3}
        const int ka = k0 + (half ? 2 : 0);
        const int kb = ka + 1;
        const int kca = ka > 48 ? 48 : ka;      // clamp: A[k>=49]==0 so value is irrelevant
        const int kcb = kb > 48 ? 48 : kb;
        v2f a, bm;
        a.x  = hasv[(size_t)(b0 + m) * CELLS + kca];
        a.y  = hasv[(size_t)(b0 + m) * CELLS + kcb];
        bm.x = Avec[ka];                        // B replicated across all N columns
        bm.y = Avec[kb];
        acc = __builtin_amdgcn_wmma_f32_16x16x4_f32(
            /*neg_a=*/false, a, /*neg_b=*/false, bm,
            /*c_mod=*/(short)0, acc, /*reuse_a=*/false, /*reuse_b=*/false);
    }

    const float cb = ws[WS_CB];
    // D layout: VGPR r, lanes 0-15 -> M=r (N=lane); lanes 16-31 -> M=r+8.
    // All N columns are equal; lanes with N==0 (lane 0 and lane 16) write out.
    if ((lane & 15) == 0) {
        const int mbase = b0 + half * 8;
#pragma unroll
        for (int r = 0; r < 8; ++r)
            loss[mbase + r] = acc[r] + cb;
    }
}

// ---------------------------------------------------------------- launch
extern "C" void kernel_launch(void* const* d_in, const int* in_sizes, int n_in,
                              void* d_out, int out_size, void* d_ws, size_t ws_size,
                              hipStream_t stream) {
    const float* out_t = (const float*)d_in[0];
    const float* tgt_t = (const float*)d_in[1];
    const int*   imw   = (const int*)d_in[2];
    const int*   imh   = (const int*)d_in[3];
    float* loss = (float*)d_out;
    float* ws   = (float*)d_ws;

    yolo_init<<<1, 256, 0, stream>>>(ws);

    // NBATCH*CELLS = 802816 = 3136 * 256 exactly
    yolo_reduce<<<3136, 256, 0, stream>>>(out_t, tgt_t, imw, imh, ws);

    yolo_finalize<<<1, 64, 0, stream>>>(ws);

    // 16384 batches / 16 per wave = 1024 waves; 8 waves (256 threads) per block
    yolo_dot_wmma<<<128, 256, 0, stream>>>(ws, loss);
}